// TripletCriterion_63187558859134
// MI455X (gfx1250) — compile-verified
//
#include <hip/hip_runtime.h>

// CDNA5 / gfx1250 triplet-margin loss using V_WMMA_F32_16X16X4_F32.
//
// Strategy: the op is L2-resident gather-bandwidth bound (16 MB batch fits the
// 192 MB L2; 1.2 GB of gather traffic). Each wave32 owns a tile of 16
// triplets; the 16 anchor.positive / anchor.negative dot products are the
// diagonal of a 16x16 WMMA accumulation over K=1024 in steps of 4.
// A-(16x4) and B-(4x16) f32 operand layouts are symmetric per lane:
//   lane L holds row (L&15), elements {k+2h, k+2h+1}, h = L>>4
// so one float2 (global_load_b64) per matrix per K-step feeds the WMMA with
// zero redundant bytes.

typedef float v2f __attribute__((ext_vector_type(2)));
typedef float v8f __attribute__((ext_vector_type(8)));

#define DIMS 1024
#define MARGIN_F 0.2f
#define WAVES_PER_BLOCK 8   // 256 threads = 8 wave32

__global__ __launch_bounds__(256)
void triplet_wmma_kernel(const float* __restrict__ batch,
                         const int*   __restrict__ triplets,  // (T,3) int32
                         float*       __restrict__ partial,   // one per wave
                         int T)
{
    const int lane       = threadIdx.x & 31;
    const int waveGlobal = blockIdx.x * WAVES_PER_BLOCK + (threadIdx.x >> 5);
    const int tileBase   = waveGlobal * 16;          // first triplet of tile

    const int row  = lane & 15;                      // triplet row in tile (M and N)
    const int half = lane >> 4;                      // which K-pair this lane supplies
    const int t    = tileBase + row;
    const int tc   = (t < T) ? t : 0;                // clamp: EXEC must stay all-1 for WMMA

    const int ia = triplets[3 * tc + 0];
    const int ip = triplets[3 * tc + 1];
    const int in_ = triplets[3 * tc + 2];

    const float* pa = batch + (size_t)ia * DIMS + 2 * half;
    const float* pp = batch + (size_t)ip * DIMS + 2 * half;
    const float* pn = batch + (size_t)in_ * DIMS + 2 * half;

    v8f acc_ap = (v8f){0.f, 0.f, 0.f, 0.f, 0.f, 0.f, 0.f, 0.f};
    v8f acc_an = (v8f){0.f, 0.f, 0.f, 0.f, 0.f, 0.f, 0.f, 0.f};

#pragma unroll 4
    for (int k = 0; k < DIMS; k += 4) {
        v2f a = *(const v2f*)(pa + k);   // A: anchors as 16x4 (rows = triplets)
        v2f p = *(const v2f*)(pp + k);   // B: positives as 4x16 (cols = triplets)
        v2f n = *(const v2f*)(pn + k);   // B: negatives as 4x16
        // 8 args: (neg_a, A, neg_b, B, c_mod, C, reuse_a, reuse_b)
        acc_ap = __builtin_amdgcn_wmma_f32_16x16x4_f32(
            false, a, false, p, (short)0, acc_ap, false, false);
        acc_an = __builtin_amdgcn_wmma_f32_16x16x4_f32(
            false, a, false, n, (short)0, acc_an, false, false);
    }

    // Diagonal extraction per documented 16x16 f32 C/D layout:
    //   diag(i), i in 0..7  -> VGPR i,   lane i
    //   diag(i), i in 8..15 -> VGPR i-8, lane i+16
    float loss = 0.f;
    int vidx = -1, myRow = -1;
    if (lane < 8)        { vidx = lane;      myRow = lane;      }
    else if (lane >= 24) { vidx = lane - 24; myRow = lane - 16; }
    if (vidx >= 0) {
        const float dap = acc_ap[vidx];
        const float dan = acc_an[vidx];
        if (tileBase + myRow < T) {
            const float l = dap - dan + MARGIN_F;
            loss = l > 0.f ? l : 0.f;
        }
    }

    // wave32 butterfly reduction of the 16 live loss values
    #pragma unroll
    for (int off = 16; off > 0; off >>= 1)
        loss += __shfl_xor(loss, off, 32);

    if (lane == 0)
        partial[waveGlobal] = loss;   // every wave slot written every call
}

__global__ __launch_bounds__(256)
void triplet_reduce_kernel(const float* __restrict__ partial, int n,
                           float invT, float* __restrict__ out)
{
    __shared__ float sm[256];
    float s = 0.f;
    for (int i = threadIdx.x; i < n; i += 256)
        s += partial[i];
    sm[threadIdx.x] = s;
    __syncthreads();
    #pragma unroll
    for (int off = 128; off > 0; off >>= 1) {
        if ((int)threadIdx.x < off) sm[threadIdx.x] += sm[threadIdx.x + off];
        __syncthreads();
    }
    if (threadIdx.x == 0)
        out[0] = sm[0] * invT;   // deterministic mean, no float atomics
}

extern "C" void kernel_launch(void* const* d_in, const int* in_sizes, int n_in,
                              void* d_out, int out_size, void* d_ws, size_t ws_size,
                              hipStream_t stream)
{
    const float* batch    = (const float*)d_in[0];
    const int*   triplets = (const int*)d_in[1];   // jax int64 -> int32 (x64 off)
    const int    T        = in_sizes[1] / 3;       // 100000

    const int nTiles   = (T + 15) / 16;                                   // 6250
    const int nBlocks  = (nTiles + WAVES_PER_BLOCK - 1) / WAVES_PER_BLOCK; // 782
    const int nPartial = nBlocks * WAVES_PER_BLOCK;                       // 6256

    float* partial = (float*)d_ws;   // nPartial * 4 bytes ~ 25 KB scratch

    triplet_wmma_kernel<<<nBlocks, 256, 0, stream>>>(batch, triplets, partial, T);
    triplet_reduce_kernel<<<1, 256, 0, stream>>>(partial, nPartial,
                                                 1.0f / (float)T, (float*)d_out);
}